// MLPPredictor_47699906789904
// MI455X (gfx1250) — compile-verified
//
#include <hip/hip_runtime.h>
#include <hip/hip_bf16.h>

// Factorized GNN edge scorer for MI455X (gfx1250, wave32).
//   Phase 1 (WMMA f32): P[n, 0:64] = h[n]*Wu^T ; P[n, 64:128] = h[n]*Wv^T
//   Phase 2 (gather):   score[e,c] = P[src[e], c] + P[dst[e], 64+c] + b[c]
// P (100000 x 128 f32 = 51.2 MB) lives in d_ws and is L2-resident (192 MB L2),
// so the edge phase's gathered reads hit L2; HBM traffic ~= the 153.6 MB
// streaming output write (nontemporal to preserve P residency).

typedef __attribute__((ext_vector_type(2))) float v2f;
typedef __attribute__((ext_vector_type(4))) float v4f;
typedef __attribute__((ext_vector_type(8))) float v8f;

#define DFEAT 128
#define NOUT  128   // 64 "src" classes + 64 "dst" classes
#define NCLS  64

// ---------------------------------------------------------------------------
// Phase 1: node projection GEMM via V_WMMA_F32_16X16X4_F32.
// Block = 256 threads = 8 waves; each wave owns one 16-row M tile and all 128
// N columns (8 accumulator fragments). K=128 -> 32 WMMA steps of K=4.
//
// LDS holds Wcat^T (Bmat[k][n], 128x128 f32) pre-swizzled into the hardware
// B-fragment layout, one ds_load_b64 per fragment:
//   entry (k4, nt, lane) stores {Bmat[k][n], Bmat[k+1][n]} where
//     k = 4*k4 + (lane>=16 ? 2 : 0),  n = nt*16 + (lane&15)
//   Bmat[k][n] = (n < 64) ? W[n][k] : W[n-64][128+k]
// ---------------------------------------------------------------------------
__global__ __launch_bounds__(256) void gnn_project_wmma(
    const float* __restrict__ h, const float* __restrict__ W,
    float* __restrict__ P, int n_nodes)
{
  __shared__ v2f ldsB[32 * 8 * 32];   // 64 KB of 320 KB WGP LDS
  const int tid = threadIdx.x;

  #pragma unroll
  for (int i = 0; i < 32; ++i) {
    const int entry = tid + (i << 8);          // 8192 float2 entries total
    const int k4    = entry >> 8;
    const int nt    = (entry >> 5) & 7;
    const int lane  = entry & 31;
    const int k     = (k4 << 2) + ((lane & 16) >> 3);   // +2 for hi half-wave
    const int n     = (nt << 4) + (lane & 15);
    const int c     = n & (NCLS - 1);
    const int col   = ((n < NCLS) ? 0 : DFEAT) + k;     // Wu cols 0:128, Wv 128:256
    const float* wp = W + (size_t)c * (2 * DFEAT) + col;
    v2f val; val.x = wp[0]; val.y = wp[1];
    ldsB[entry] = val;
  }
  __syncthreads();

  const int wave  = tid >> 5;
  const int lane  = tid & 31;
  const int mtile = blockIdx.x * 8 + wave;
  const int row0  = mtile * 16;
  if (row0 >= n_nodes) return;                 // whole-wave exit: EXEC stays all-1s

  // A fragment (16x4 f32): lane l<16 -> row l, K+0/K+1 ; lanes 16-31 -> K+2/K+3
  int arow = row0 + (lane & 15);
  if (arow >= n_nodes) arow = n_nodes - 1;     // clamp reads for ragged tail
  const float* aptr = h + (size_t)arow * DFEAT + ((lane & 16) >> 3);

  v8f acc[8] = {};                             // 16x128 f32 tile, zero-initialized

  #pragma unroll 2
  for (int k4 = 0; k4 < 32; ++k4) {
    const v2f aF = *(const v2f*)(aptr + (k4 << 2));
    const v2f* bbase = &ldsB[(k4 << 8) + lane];
    #pragma unroll
    for (int nt = 0; nt < 8; ++nt) {
      const v2f bF = bbase[nt << 5];
      // (neg_a, A, neg_b, B, c_mod, C, reuse_a, reuse_b)
      acc[nt] = __builtin_amdgcn_wmma_f32_16x16x4_f32(
          false, aF, false, bF, (short)0, acc[nt], false, false);
    }
  }

  // C/D layout: VGPR r, lanes 0-15 -> M=r ; lanes 16-31 -> M=8+r ; N = lane&15
  const int rbase = row0 + ((lane & 16) >> 1); // +8 for hi half-wave
  float* prow = P + (size_t)rbase * NOUT + (lane & 15);
  if (row0 + 16 <= n_nodes) {
    // Full tile (the only case when n_nodes % 16 == 0): straight-line stores,
    // no per-element EXEC gymnastics.
    #pragma unroll
    for (int nt = 0; nt < 8; ++nt) {
      #pragma unroll
      for (int r = 0; r < 8; ++r)
        prow[(size_t)r * NOUT + nt * 16] = acc[nt][r];
    }
  } else {
    // Ragged tail tile: guard each row.
    #pragma unroll
    for (int nt = 0; nt < 8; ++nt) {
      #pragma unroll
      for (int r = 0; r < 8; ++r)
        if (rbase + r < n_nodes)
          prow[(size_t)r * NOUT + nt * 16] = acc[nt][r];
    }
  }
}

// ---------------------------------------------------------------------------
// Phase 2: per-edge gather-add. Two edges per wave: lanes 0-15 -> edge 2w,
// lanes 16-31 -> edge 2w+1; each lane handles 4 classes as float4, so every
// stream moves through b128 vmem ops (coalesced 256 B per half-wave).
// Output uses nontemporal stores so the 153.6 MB stream does not evict the
// L2-resident P table.
// ---------------------------------------------------------------------------
__global__ __launch_bounds__(256) void gnn_edge_gather(
    const float* __restrict__ P, const int* __restrict__ src,
    const int* __restrict__ dst, const float* __restrict__ b,
    float* __restrict__ out, int n_edges)
{
  const int gid  = blockIdx.x * 256 + threadIdx.x;
  const int lane = threadIdx.x & 31;
  const int e    = ((gid >> 5) << 1) + (lane >> 4);
  if (e >= n_edges) return;
  const int c4 = (lane & 15) << 2;             // class offset: 0,4,...,60
  const int s = src[e];
  const int d = dst[e];
  const v4f ps = *(const v4f*)(P + (size_t)s * NOUT + c4);
  const v4f pd = *(const v4f*)(P + (size_t)d * NOUT + NCLS + c4);
  const v4f vb = *(const v4f*)(b + c4);
  const v4f r  = ps + pd + vb;
  __builtin_nontemporal_store(r, (v4f*)(out + (size_t)e * NCLS + c4));
}

// ---------------------------------------------------------------------------
// Fallback if workspace can't hold P: direct per-edge scoring (LDS-staged
// feature rows, one block per edge, one thread per class).
// ---------------------------------------------------------------------------
__global__ __launch_bounds__(64) void gnn_direct_fallback(
    const float* __restrict__ h, const int* __restrict__ src,
    const int* __restrict__ dst, const float* __restrict__ W,
    const float* __restrict__ b, float* __restrict__ out, int n_edges)
{
  __shared__ float hs[DFEAT];
  __shared__ float hd[DFEAT];
  const int e = blockIdx.x;
  if (e >= n_edges) return;
  const int c = threadIdx.x;
  const int s = src[e];
  const int d = dst[e];
  for (int i = c; i < DFEAT; i += 64) {
    hs[i] = h[(size_t)s * DFEAT + i];
    hd[i] = h[(size_t)d * DFEAT + i];
  }
  __syncthreads();
  const float* wr = W + (size_t)c * (2 * DFEAT);
  float acc = b[c];
  #pragma unroll 8
  for (int i = 0; i < DFEAT; ++i)
    acc = fmaf(wr[i], hs[i], fmaf(wr[DFEAT + i], hd[i], acc));
  out[(size_t)e * NCLS + c] = acc;
}

extern "C" void kernel_launch(void* const* d_in, const int* in_sizes, int n_in,
                              void* d_out, int out_size, void* d_ws, size_t ws_size,
                              hipStream_t stream)
{
  // setup_inputs order: h, src, dst, W, b
  const float* h   = (const float*)d_in[0];
  const int*   src = (const int*)d_in[1];   // JAX default (x64 off) => int32
  const int*   dst = (const int*)d_in[2];
  const float* W   = (const float*)d_in[3];
  const float* b   = (const float*)d_in[4];
  float* out = (float*)d_out;

  const int n_nodes = in_sizes[0] / DFEAT;
  const int n_edges = in_sizes[1];

  const size_t p_bytes = (size_t)n_nodes * NOUT * sizeof(float);
  if (ws_size >= p_bytes) {
    float* P = (float*)d_ws;
    const int mtiles = (n_nodes + 15) / 16;
    const int blocks = (mtiles + 7) / 8;
    gnn_project_wmma<<<blocks, 256, 0, stream>>>(h, W, P, n_nodes);

    // 16 edges per 256-thread block (2 edges per wave)
    const int gblocks = (n_edges + 15) / 16;
    gnn_edge_gather<<<gblocks, 256, 0, stream>>>(P, src, dst, b, out, n_edges);
  } else {
    gnn_direct_fallback<<<n_edges, 64, 0, stream>>>(h, src, dst, W, b, out, n_edges);
  }
}